// Embedder_83141976916519
// MI455X (gfx1250) — compile-verified
//
#include <hip/hip_runtime.h>
#include <stdint.h>

typedef __attribute__((ext_vector_type(2))) float v2f;
typedef __attribute__((ext_vector_type(4))) float v4f;
typedef __attribute__((ext_vector_type(8))) float v8f;

#define LEVELS 16
#define FEAT 16
#define TSZ 262147
#define START_HASH 11
#define OUTW 19

__constant__ int c_res[LEVELS] = {2,2,3,5,7,10,13,19,26,36,50,69,95,131,181,250};
__constant__ float c_size[LEVELS] = {
    (float)(1.0/1.0),   (float)(1.0/1.0),   (float)(1.0/2.0),   (float)(1.0/4.0),
    (float)(1.0/6.0),   (float)(1.0/9.0),   (float)(1.0/12.0),  (float)(1.0/18.0),
    (float)(1.0/25.0),  (float)(1.0/35.0),  (float)(1.0/49.0),  (float)(1.0/68.0),
    (float)(1.0/94.0),  (float)(1.0/130.0), (float)(1.0/180.0), (float)(1.0/249.0)};

__device__ __forceinline__ int clampi(int v, int lo, int hi) {
    return v < lo ? lo : (v > hi ? hi : v);
}

// ---------------------------------------------------------------------------
// Kernel A: tsum[r] = sum_f table[r*16 + f] via V_WMMA_F32_16X16X4_F32.
// One wave handles 16 consecutive rows. A-matrix (16x4 f32) layout:
//   lanes 0-15  : M=lane,    VGPR0=K0, VGPR1=K1
//   lanes 16-31 : M=lane-16, VGPR0=K2, VGPR1=K3
// -> per chunk c, lane loads b64 at row*64 + (lane>=16 ? 8 : 0) + c*16 bytes.
// B = all-ones 4x16. After 4 accumulating WMMAs, D[m,n] = rowsum(m) for all n.
// D layout: VGPR j = rows {j (lanes 0-15), j+8 (lanes 16-31)}, so lane 0 holds
// rowsums 0..7 and lane 16 holds rowsums 8..15 -> two b128 stores each.
// ---------------------------------------------------------------------------
__global__ void reduce_rows_wmma(const float* __restrict__ table,
                                 float* __restrict__ tsum,
                                 long nrows) {
    const int wavesPerBlock = blockDim.x >> 5;
    long wave = (long)blockIdx.x * wavesPerBlock + (threadIdx.x >> 5);
    long row0 = wave * 16;
    if (row0 >= nrows) return;                 // wave-uniform: EXEC stays all-1s

    int lane = threadIdx.x & 31;
    int m    = lane & 15;
    int kh   = lane >> 4;                      // 0: K=0..1, 1: K=2..3

    const float* src = table + (row0 + m) * (long)FEAT + kh * 2;

    v2f ones = {1.0f, 1.0f};
    v8f acc = {};
#pragma unroll
    for (int c = 0; c < 4; ++c) {
        v2f a = *(const v2f*)(src + c * 4);
        acc = __builtin_amdgcn_wmma_f32_16x16x4_f32(
            /*neg_a=*/false, a, /*neg_b=*/false, ones,
            /*c_mod=*/(short)0, acc, /*reuse_a=*/false, /*reuse_b=*/false);
    }

    if (m == 0) {                              // lanes 0 and 16 write results
        float* dst = tsum + row0 + kh * 8;
        v4f lo = {acc[0], acc[1], acc[2], acc[3]};
        v4f hi = {acc[4], acc[5], acc[6], acc[7]};
        *(v4f*)(dst)     = lo;
        *(v4f*)(dst + 4) = hi;
    }
}

// ---------------------------------------------------------------------------
// Corner index + trilinear weight setup (replicates reference exactly).
// ---------------------------------------------------------------------------
__device__ __forceinline__ void corner_setup(float fx, float fy, float fz,
                                             int res, bool hash,
                                             uint32_t idx[8], float w[8]) {
    int rm1 = res - 1;
    int c0x = clampi((int)fx, 0, rm1);
    int c0y = clampi((int)fy, 0, rm1);
    int c0z = clampi((int)fz, 0, rm1);
    int c1x = clampi((int)(fx + 1.0f), 0, rm1);
    int c1y = clampi((int)(fy + 1.0f), 0, rm1);
    int c1z = clampi((int)(fz + 1.0f), 0, rm1);
    float ox = fx - (float)c0x;
    float oy = fy - (float)c0y;
    float oz = fz - (float)c0z;

    int   cx_[2] = {c0x, c1x}, cy_[2] = {c0y, c1y}, cz_[2] = {c0z, c1z};
    float wx_[2] = {1.0f - ox, ox};
    float wy_[2] = {1.0f - oy, oy};
    float wz_[2] = {1.0f - oz, oz};

    int r2 = res * res;
#pragma unroll
    for (int k = 0; k < 8; ++k) {
        int bx = (k >> 2) & 1, by = (k >> 1) & 1, bz = k & 1;
        int cx = cx_[bx], cy = cy_[by], cz = cz_[bz];
        w[k] = wx_[bx] * wy_[by] * wz_[bz];
        if (hash) {
            uint64_t h = (uint64_t)(uint32_t)cx
                       ^ ((uint64_t)(uint32_t)cy * 2654435761ull)
                       ^ ((uint64_t)(uint32_t)cz * 805459861ull);
            idx[k] = (uint32_t)(h % (uint64_t)TSZ);
        } else {
            idx[k] = (uint32_t)(cx * r2 + cy * res + cz);
        }
    }
}

// ---------------------------------------------------------------------------
// Kernel B: one thread per (point, level); level = blockIdx.y (wave-uniform).
// Gathers 8 scalars from the precomputed row-sum table (L2-resident, 16.8MB).
// ---------------------------------------------------------------------------
__global__ void embed_from_sums(const float* __restrict__ xyz,
                                const float* __restrict__ tsum,
                                const float* __restrict__ bounds,
                                float* __restrict__ out, int npts) {
    int p = blockIdx.x * blockDim.x + threadIdx.x;
    int l = blockIdx.y;
    if (p >= npts) return;

    float x = (xyz[3 * p + 0] - bounds[0]) / (bounds[3] - bounds[0]);
    float y = (xyz[3 * p + 1] - bounds[1]) / (bounds[4] - bounds[1]);
    float z = (xyz[3 * p + 2] - bounds[2]) / (bounds[5] - bounds[2]);

    int   res = c_res[l];
    float sz  = c_size[l];
    float fx = x / sz, fy = y / sz, fz = z / sz;

    uint32_t idx[8]; float w[8];
    corner_setup(fx, fy, fz, res, l >= START_HASH, idx, w);

    const float* tl = tsum + (long)l * TSZ;
    float v = 0.0f;
#pragma unroll
    for (int k = 0; k < 8; ++k) v += w[k] * tl[idx[k]];

    long ob = (long)p * OUTW;
    out[ob + 3 + l] = v;
    if (l == 0) { out[ob] = x; out[ob + 1] = y; out[ob + 2] = z; }
}

// ---------------------------------------------------------------------------
// Fallback (no workspace): gather full 16-float rows and reduce inline.
// ---------------------------------------------------------------------------
__global__ void embed_direct(const float* __restrict__ xyz,
                             const float* __restrict__ table,
                             const float* __restrict__ bounds,
                             float* __restrict__ out, int npts) {
    int p = blockIdx.x * blockDim.x + threadIdx.x;
    int l = blockIdx.y;
    if (p >= npts) return;

    float x = (xyz[3 * p + 0] - bounds[0]) / (bounds[3] - bounds[0]);
    float y = (xyz[3 * p + 1] - bounds[1]) / (bounds[4] - bounds[1]);
    float z = (xyz[3 * p + 2] - bounds[2]) / (bounds[5] - bounds[2]);

    int   res = c_res[l];
    float sz  = c_size[l];
    float fx = x / sz, fy = y / sz, fz = z / sz;

    uint32_t idx[8]; float w[8];
    corner_setup(fx, fy, fz, res, l >= START_HASH, idx, w);

    const float* tl = table + (long)l * TSZ * FEAT;
    float v = 0.0f;
#pragma unroll
    for (int k = 0; k < 8; ++k) {
        const float* row = tl + (long)idx[k] * FEAT;
        v4f a = *(const v4f*)(row);
        v4f b = *(const v4f*)(row + 4);
        v4f c = *(const v4f*)(row + 8);
        v4f d = *(const v4f*)(row + 12);
        float s = ((a[0] + a[1]) + (a[2] + a[3])) + ((b[0] + b[1]) + (b[2] + b[3]))
                + ((c[0] + c[1]) + (c[2] + c[3])) + ((d[0] + d[1]) + (d[2] + d[3]));
        v += w[k] * s;
    }

    long ob = (long)p * OUTW;
    out[ob + 3 + l] = v;
    if (l == 0) { out[ob] = x; out[ob + 1] = y; out[ob + 2] = z; }
}

extern "C" void kernel_launch(void* const* d_in, const int* in_sizes, int n_in,
                              void* d_out, int out_size, void* d_ws, size_t ws_size,
                              hipStream_t stream) {
    const float* xyz    = (const float*)d_in[0];
    const float* table  = (const float*)d_in[1];
    const float* bounds = (const float*)d_in[2];
    float* out = (float*)d_out;

    int npts = in_sizes[0] / 3;
    long nrows = (long)LEVELS * TSZ;              // 4,194,352 rows (16 floats each)
    size_t need = (size_t)nrows * sizeof(float);  // 16.8 MB row-sum table

    dim3 blockB(256);
    dim3 gridB((npts + 255) / 256, LEVELS);

    if (ws_size >= need && d_ws != nullptr) {
        float* tsum = (float*)d_ws;
        // One wave per 16-row tile; 8 waves per 256-thread block.
        long tiles = nrows / 16;                  // exact: L divisible by 16
        int blocksA = (int)((tiles + 7) / 8);
        reduce_rows_wmma<<<blocksA, 256, 0, stream>>>(table, tsum, nrows);
        embed_from_sums<<<gridB, blockB, 0, stream>>>(xyz, tsum, bounds, out, npts);
    } else {
        embed_direct<<<gridB, blockB, 0, stream>>>(xyz, table, bounds, out, npts);
    }
}